// DCP_LOSS_33019708571839
// MI455X (gfx1250) — compile-verified
//
#include <hip/hip_runtime.h>
#include <math.h>

#define B_      8
#define H_      1024
#define W_      1024
#define HW_     (H_ * W_)
#define MASKN   (B_ * HW_)
#define NBINS   20
#define SPER    36                      // 16 block sums + 20 histo bins
#define NSTAT   (2 * 2 * B_ * SPER)     // sides * stains * images * SPER = 1152

typedef float v2f __attribute__((ext_vector_type(2)));
typedef float v8f __attribute__((ext_vector_type(8)));

// Raw CDNA5 hardware transcendentals (v_log_f32 = log2, v_exp_f32 = exp2).
#define LOG2F(x) __builtin_amdgcn_logf(x)
#define EXP2F(x) __builtin_amdgcn_exp2f(x)

struct Consts {
    float wh0, wh1, wh2;   // HED_FROM_RGB[:,0]  (H stain weights)
    float wd0, wd1, wd2;   // HED_FROM_RGB[:,2]  (D stain weights)
    float kh0, kh1, kh2;   // exp2 coefficients: LOG_ADJ * RGB_FROM_HED[0][c] * log2(e)
    float kd0, kd1, kd2;   // LOG_ADJ * RGB_FROM_HED[2][c] * log2(e)
    float clg;             // ln2 / LOG_ADJ   (log2(rgb)*clg == log(rgb)/LOG_ADJ)
    float cal;             // 10^(-e)
    float fodc;            // -log10(2)       (log2(g+cal)*fodc == -log10(g+cal))
    float invbw;           // 20/e            (histogram bin scale)
};

__global__ void dcp_zero_kernel(float* __restrict__ stats) {
    for (int i = threadIdx.x; i < NSTAT; i += 256) stats[i] = 0.0f;
}

// grid (4, 128, 16): x = 256-wide column chunk, y = 8-row band, z = side*8 + image
__global__ __launch_bounds__(256) void dcp_pixel_kernel(
    const float* __restrict__ inp, const float* __restrict__ tgt,
    float* __restrict__ out, float* __restrict__ stats, Consts K)
{
    __shared__ float hist[2][NBINS];
    __shared__ float red[2][256];

    const int tid  = threadIdx.x;
    const int z    = blockIdx.z;
    const int side = z >> 3;
    const int b    = z & 7;
    const int w0   = blockIdx.x << 8;
    const int h0   = blockIdx.y << 3;

    if (tid < 2 * NBINS) ((float*)hist)[tid] = 0.0f;
    __syncthreads();

    const float* __restrict__ src = (side ? tgt : inp) + (size_t)b * 3 * HW_;
    float* __restrict__ mh = out + 1 + (size_t)side       * MASKN + (size_t)b * HW_;
    float* __restrict__ md = out + 1 + (size_t)(2 + side) * MASKN + (size_t)b * HW_;

    float ph = 0.0f, pd = 0.0f;

    #pragma unroll
    for (int r = 0; r < 8; ++r) {
        const int p = (h0 + r) * W_ + w0 + tid;
        float cr = fmaxf(src[p],           1e-6f);
        float cg = fmaxf(src[HW_ + p],     1e-6f);
        float cb = fmaxf(src[2 * HW_ + p], 1e-6f);
        float lr = LOG2F(cr) * K.clg;    // log(rgb)/LOG_ADJ
        float lg = LOG2F(cg) * K.clg;
        float lb = LOG2F(cb) * K.clg;
        float sh = fmaxf(lr * K.wh0 + lg * K.wh1 + lb * K.wh2, 0.0f);
        float sd = fmaxf(lr * K.wd0 + lg * K.wd1 + lb * K.wd2, 0.0f);
        // ---- H stain ----
        {
            float g = 0.2125f * EXP2F(K.kh0 * sh)
                    + 0.7154f * EXP2F(K.kh1 * sh)
                    + 0.0721f * EXP2F(K.kh2 * sh);
            g = fminf(g, 1.0f);
            float fod = fmaxf(LOG2F(g + K.cal) * K.fodc, 0.0f);
            ph += (fod < 0.15f) ? 0.0f : fod;
            mh[p] = (fod >= 0.3f) ? 1.0f : 0.0f;
            int bin = (int)fminf(fod * K.invbw, 19.0f);
            atomicAdd(&hist[0][bin], fod);
        }
        // ---- D stain ----
        {
            float g = 0.2125f * EXP2F(K.kd0 * sd)
                    + 0.7154f * EXP2F(K.kd1 * sd)
                    + 0.0721f * EXP2F(K.kd2 * sd);
            g = fminf(g, 1.0f);
            float fod = fmaxf(LOG2F(g + K.cal) * K.fodc, 0.0f);
            pd += (fod < 0.15f) ? 0.0f : fod;
            md[p] = (fod >= 0.3f) ? 1.0f : 0.0f;
            int bin = (int)fminf(fod * K.invbw, 19.0f);
            atomicAdd(&hist[1][bin], fod);
        }
    }

    red[0][tid] = ph;
    red[1][tid] = pd;
    __syncthreads();

    const int baseh = ((side * 2 + 0) * B_ + b) * SPER;
    const int based = ((side * 2 + 1) * B_ + b) * SPER;
    const int blk   = ((h0 >> 8) << 2) + blockIdx.x;   // (h/256)*4 + (w/256)

    if (tid < 32) {
        // WMMA reduction: D += ones(16x4) x B(4x16); each wmma folds 64 LDS
        // partials into 16 replicated column sums; 4 chained per stain.
        v2f ones; ones[0] = 1.0f; ones[1] = 1.0f;
        v8f acch = {};
        #pragma unroll
        for (int i = 0; i < 4; ++i) {
            v2f bm; bm[0] = red[0][i * 64 + tid]; bm[1] = red[0][i * 64 + 32 + tid];
            acch = __builtin_amdgcn_wmma_f32_16x16x4_f32(
                false, ones, false, bm, (short)0, acch, false, false);
        }
        v8f accd = {};
        #pragma unroll
        for (int i = 0; i < 4; ++i) {
            v2f bm; bm[0] = red[1][i * 64 + tid]; bm[1] = red[1][i * 64 + 32 + tid];
            accd = __builtin_amdgcn_wmma_f32_16x16x4_f32(
                false, ones, false, bm, (short)0, accd, false, false);
        }
        float th = acch[0], td = accd[0];   // lane n holds colsum[n%16]
        th += __shfl_xor(th, 1, 32); td += __shfl_xor(td, 1, 32);
        th += __shfl_xor(th, 2, 32); td += __shfl_xor(td, 2, 32);
        th += __shfl_xor(th, 4, 32); td += __shfl_xor(td, 4, 32);
        th += __shfl_xor(th, 8, 32); td += __shfl_xor(td, 8, 32);
        if (tid == 0) {
            atomicAdd(&stats[baseh + blk], th);
            atomicAdd(&stats[based + blk], td);
        }
    }
    if (tid >= 32 && tid < 32 + 2 * NBINS) {
        int t = tid - 32;
        int st = t / NBINS, bin = t % NBINS;
        atomicAdd(&stats[((side * 2 + st) * B_ + b) * SPER + 16 + bin],
                  hist[st][bin]);
    }
}

__global__ void dcp_finalize_kernel(const float* __restrict__ stats,
                                    float* __restrict__ out)
{
    const int lane = threadIdx.x;  // 32 threads, one wave
    float contrib = 0.0f;
    if (lane < 16) {
        const int t = lane >> 3;       // 0=H, 1=D
        const int b = lane & 7;
        const float invHW = 1.0f / (float)HW_;
        const float sc    = 16.0f * invHW;     // 1/(HW/16)
        const float* si = stats + ((0 * 2 + t) * B_ + b) * SPER;
        const float* st = stats + ((1 * 2 + t) * B_ + b) * SPER;
        float ia = 0.0f, ta = 0.0f, bs = 0.0f;
        for (int k = 0; k < 16; ++k) {
            float x = si[k], y = st[k];
            ia += x; ta += y;
            float d = (x - y) * sc;
            bs += d * d;
        }
        float hs = 0.0f;
        for (int k = 0; k < NBINS; ++k) {
            float d = (si[16 + k] - st[16 + k]) * invHW;
            hs += d * d;
        }
        float his_t = hs * (1.0f / (float)B_);
        float diff  = ia - ta;
        float dd    = diff * invHW;
        float avg_t = dd * dd;
        bool cond = (diff >= ta * -0.4f) && (diff <= ta * 0.4f);
        float term = cond ? his_t : (avg_t + his_t);
        contrib = term + bs * (1.0f / 128.0f);   // blk_t = mean over B*16
    }
    contrib += __shfl_xor(contrib, 16, 32);
    contrib += __shfl_xor(contrib, 8, 32);
    contrib += __shfl_xor(contrib, 4, 32);
    contrib += __shfl_xor(contrib, 2, 32);
    contrib += __shfl_xor(contrib, 1, 32);
    if (lane == 0) out[0] = contrib;
}

extern "C" void kernel_launch(void* const* d_in, const int* in_sizes, int n_in,
                              void* d_out, int out_size, void* d_ws, size_t ws_size,
                              hipStream_t stream) {
    const float* inp = (const float*)d_in[0];
    const float* tgt = (const float*)d_in[1];
    float* out   = (float*)d_out;
    float* stats = (float*)d_ws;

    // Host-side double-precision constants (matches reference's f64 inverse).
    const double M[3][3] = {{0.65, 0.70, 0.29},
                            {0.07, 0.99, 0.11},
                            {0.27, 0.57, 0.78}};
    double c00 =  M[1][1]*M[2][2] - M[1][2]*M[2][1];
    double c01 = -(M[1][0]*M[2][2] - M[1][2]*M[2][0]);
    double c02 =  M[1][0]*M[2][1] - M[1][1]*M[2][0];
    double det = M[0][0]*c00 + M[0][1]*c01 + M[0][2]*c02;
    double inv[3][3];
    inv[0][0] = c00 / det;
    inv[1][0] = c01 / det;
    inv[2][0] = c02 / det;
    inv[0][1] = -(M[0][1]*M[2][2] - M[0][2]*M[2][1]) / det;
    inv[1][1] =  (M[0][0]*M[2][2] - M[0][2]*M[2][0]) / det;
    inv[2][1] = -(M[0][0]*M[2][1] - M[0][1]*M[2][0]) / det;
    inv[0][2] =  (M[0][1]*M[1][2] - M[0][2]*M[1][1]) / det;
    inv[1][2] = -(M[0][0]*M[1][2] - M[0][2]*M[1][0]) / det;
    inv[2][2] =  (M[0][0]*M[1][1] - M[0][1]*M[1][0]) / det;

    const double LOG_ADJ = log(1e-6);
    const double LOG2E   = 1.4426950408889634074;

    Consts K;
    K.wh0 = (float)inv[0][0]; K.wh1 = (float)inv[1][0]; K.wh2 = (float)inv[2][0];
    K.wd0 = (float)inv[0][2]; K.wd1 = (float)inv[1][2]; K.wd2 = (float)inv[2][2];
    K.kh0 = (float)(LOG_ADJ * M[0][0] * LOG2E);
    K.kh1 = (float)(LOG_ADJ * M[0][1] * LOG2E);
    K.kh2 = (float)(LOG_ADJ * M[0][2] * LOG2E);
    K.kd0 = (float)(LOG_ADJ * M[2][0] * LOG2E);
    K.kd1 = (float)(LOG_ADJ * M[2][1] * LOG2E);
    K.kd2 = (float)(LOG_ADJ * M[2][2] * LOG2E);
    K.clg   = (float)(M_LN2 / LOG_ADJ);
    K.cal   = (float)pow(10.0, -exp(1.0));
    K.fodc  = (float)(-log10(2.0));
    K.invbw = (float)(20.0 / M_E);

    dcp_zero_kernel<<<1, 256, 0, stream>>>(stats);
    dcp_pixel_kernel<<<dim3(4, 128, 16), 256, 0, stream>>>(inp, tgt, out, stats, K);
    dcp_finalize_kernel<<<1, 32, 0, stream>>>(stats, out);
}